// SAGE_15118284882235
// MI455X (gfx1250) — compile-verified
//
#include <hip/hip_runtime.h>
#include <math.h>

// ---------------------------------------------------------------------------
// GraphSAGE 3-layer forward for MI455X (gfx1250), wave32, WMMA f32 16x16x4.
// One wave owns a 16-row strip and accumulates NT 16-col tiles at once.
// B fragments for all NT tiles are loaded as a batch (single base pointer +
// immediate offsets) before the NT WMMAs issue, so load latency overlaps the
// matrix pipe instead of serializing load->wait->wmma.
// ---------------------------------------------------------------------------

typedef __attribute__((ext_vector_type(2))) float v2f;
typedef __attribute__((ext_vector_type(8))) float v8f;

// ---------------- problem constants (from the reference) -------------------
#define N_SRC0 512000
#define N_DST0 102400
#define N_DST1 10240
#define N_DST2 1024
#define E0 1024000
#define E1 102400
#define E2 10240
#define IN_C 100
#define HID_C 256
#define OUT_C 47

// ---------------- workspace layout (bytes) ---------------------------------
// h1   : 102400*256*4 = 104,857,600
// agg0 : 102400*100*4 =  40,960,000   (reused: agg1 @ same off, h2 after it)
// cnt  :       409,600                 (cnt0; reused for cnt1/cnt2)
#define H1_OFF   ((size_t)0)
#define AGG0_OFF ((size_t)104857600)
#define CNT_OFF  ((size_t)145817600)
#define AGG1_OFF AGG0_OFF
#define H2_OFF   (AGG0_OFF + (size_t)10485760)   // 10240*256*4
#define AGG2_OFF AGG0_OFF

// ---------------------------------------------------------------------------
__global__ __launch_bounds__(256) void sage_zero_f32(float* __restrict__ p, int n) {
    int i = blockIdx.x * blockDim.x + threadIdx.x;
    if (i < n) p[i] = 0.0f;
}

// One wave per edge: lanes stride over C features, atomic-add into agg[dst],
// lane 0 bumps the float edge count for the mean.
__global__ __launch_bounds__(256) void sage_scatter(
    const float* __restrict__ x, const int* __restrict__ src,
    const int* __restrict__ dst, float* __restrict__ agg,
    float* __restrict__ cnt, int n_edges, int C)
{
    int wave = (blockIdx.x * blockDim.x + threadIdx.x) >> 5;
    int lane = threadIdx.x & 31;
    if (wave >= n_edges) return;
    int s = src[wave];
    int d = dst[wave];
    const float* xs = x + (size_t)s * C;
    float* ad = agg + (size_t)d * C;
    for (int c = lane; c < C; c += 32)
        atomicAdd(&ad[c], xs[c]);
    if (lane == 0)
        atomicAdd(&cnt[d], 1.0f);
}

// agg[r][c] /= max(cnt[r], 1)  (in place -> mean)
__global__ __launch_bounds__(256) void sage_meandiv(
    float* __restrict__ agg, const float* __restrict__ cnt, int total, int C)
{
    int i = blockIdx.x * blockDim.x + threadIdx.x;
    if (i < total) {
        float c = cnt[i / C];
        agg[i] = agg[i] / fmaxf(c, 1.0f);
    }
}

// out[n_rows x n_out] = A1 @ Wl^T + bl + A2 @ Wr^T   (Wl/Wr are [n_out x C])
// One wave computes a 16 x (NT*16) strip with V_WMMA_F32_16X16X4_F32,
// running two K-passes (A1 x Wl, A2 x Wr) into NT f32 accumulators.
//
// f32 WMMA VGPR layouts (ISA 7.12.2):
//   A 16x4 : lanes 0-15 -> (M=lane, K=0 in v0, K=1 in v1)
//            lanes 16-31 -> (M=lane-16, K=2 in v0, K=3 in v1)
//   B 4x16 : lanes 0-15 -> (N=lane, K=0/1), lanes 16-31 -> (N=lane-16, K=2/3)
//   C/D    : v8f, row = r + 8*(lane>=16), col = lane%16
// BOUND=true handles n_out not divisible by 16 with clamped pointer + 0/1
// mask so EXEC stays all-ones through every v_wmma (ISA requirement).
// All fragment addresses are 8-byte aligned (C is a multiple of 4, ka even).
template <int NT, bool BOUND>
__global__ __launch_bounds__(256) void sage_gemm_wmma(
    const float* __restrict__ A1, const float* __restrict__ A2,
    const float* __restrict__ Wl, const float* __restrict__ Wr,
    const float* __restrict__ bl, float* __restrict__ out,
    int n_rows, int C, int n_out, int groupsN, int do_relu)
{
    int wave = (blockIdx.x * blockDim.x + threadIdx.x) >> 5;
    int lane = threadIdx.x & 31;
    int tileM = wave / groupsN;
    int group = wave - tileM * groupsN;
    if (tileM * 16 >= n_rows) return;        // wave-uniform exit; EXEC stays full

    int halfSel = lane >> 4;                 // 0: lanes 0-15, 1: lanes 16-31
    int l16 = lane & 15;
    int rowA = tileM * 16 + l16;             // always valid (n_rows % 16 == 0)
    int colB0 = group * NT * 16 + l16;       // tile t's column = colB0 + t*16

    int   colBc[NT];                         // clamped weight row (BOUND only)
    float mask[NT];
    #pragma unroll
    for (int t = 0; t < NT; ++t) {
        int c = colB0 + t * 16;
        colBc[t] = BOUND ? (c < n_out ? c : (n_out - 1)) : c;
        mask[t]  = BOUND ? ((c < n_out) ? 1.0f : 0.0f) : 1.0f;
    }

    v8f acc[NT];
    #pragma unroll
    for (int t = 0; t < NT; ++t)
        acc[t] = (v8f){0.f, 0.f, 0.f, 0.f, 0.f, 0.f, 0.f, 0.f};

    for (int pass = 0; pass < 2; ++pass) {
        const float* A = pass ? A2 : A1;
        const float* W = pass ? Wr : Wl;
        const float* arow  = A + (size_t)rowA * C;
        const float* wbase = W + (size_t)colB0 * C;   // unbounded: + t*16*C
        for (int k = 0; k < C; k += 4) {
            int ka = k + 2 * halfSel;        // this lane supplies K = ka, ka+1
            v2f a = *(const v2f*)(arow + ka);
            // ---- batch all NT B-fragment loads (overlap their latency) ----
            v2f b[NT];
            #pragma unroll
            for (int t = 0; t < NT; ++t) {
                if (BOUND) {
                    v2f w = *(const v2f*)(W + (size_t)colBc[t] * C + ka);
                    b[t].x = w.x * mask[t];
                    b[t].y = w.y * mask[t];
                } else {
                    // constant byte offset t*16*C*4 folds into the load imm
                    b[t] = *(const v2f*)(wbase + (size_t)t * 16 * C + ka);
                }
            }
            // ---- then issue NT WMMAs back-to-back -------------------------
            #pragma unroll
            for (int t = 0; t < NT; ++t)
                acc[t] = __builtin_amdgcn_wmma_f32_16x16x4_f32(
                    false, a, false, b[t], (short)0, acc[t], false, false);
        }
    }

    #pragma unroll
    for (int t = 0; t < NT; ++t) {
        int colB = colB0 + t * 16;
        if (!BOUND || colB < n_out) {
            float bias = bl[colB];
            #pragma unroll
            for (int r = 0; r < 8; ++r) {
                int row = tileM * 16 + r + 8 * halfSel;
                float v = acc[t][r] + bias;
                if (do_relu) v = fmaxf(v, 0.0f);
                out[(size_t)row * n_out + colB] = v;
            }
        }
    }
}

// In-place row-wise log_softmax; one wave32 per row (n_out <= 64).
__global__ __launch_bounds__(256) void sage_logsoftmax(
    float* __restrict__ out, int n_rows, int n_out)
{
    int wave = (blockIdx.x * blockDim.x + threadIdx.x) >> 5;
    int lane = threadIdx.x & 31;
    if (wave >= n_rows) return;
    float* row = out + (size_t)wave * n_out;
    float v0 = (lane < n_out) ? row[lane] : -INFINITY;
    float v1 = (lane + 32 < n_out) ? row[lane + 32] : -INFINITY;
    float m = fmaxf(v0, v1);
    #pragma unroll
    for (int off = 16; off > 0; off >>= 1)
        m = fmaxf(m, __shfl_xor(m, off, 32));
    float e0 = (lane < n_out) ? expf(v0 - m) : 0.0f;
    float e1 = (lane + 32 < n_out) ? expf(v1 - m) : 0.0f;
    float s = e0 + e1;
    #pragma unroll
    for (int off = 16; off > 0; off >>= 1)
        s += __shfl_xor(s, off, 32);
    float lse = m + logf(s);
    if (lane < n_out) row[lane] = v0 - lse;
    if (lane + 32 < n_out) row[lane + 32] = v1 - lse;
}

// ---------------------------------------------------------------------------
static inline int divup(long long a, int b) { return (int)((a + b - 1) / b); }

extern "C" void kernel_launch(void* const* d_in, const int* in_sizes, int n_in,
                              void* d_out, int out_size, void* d_ws, size_t ws_size,
                              hipStream_t stream) {
    (void)in_sizes; (void)n_in; (void)out_size; (void)ws_size;

    const float* x    = (const float*)d_in[0];
    const int*   src0 = (const int*)  d_in[1];
    const int*   dst0 = (const int*)  d_in[2];
    const int*   src1 = (const int*)  d_in[3];
    const int*   dst1 = (const int*)  d_in[4];
    const int*   src2 = (const int*)  d_in[5];
    const int*   dst2 = (const int*)  d_in[6];
    const float* Wl0  = (const float*)d_in[7];
    const float* bl0  = (const float*)d_in[8];
    const float* Wr0  = (const float*)d_in[9];
    const float* Wl1  = (const float*)d_in[10];
    const float* bl1  = (const float*)d_in[11];
    const float* Wr1  = (const float*)d_in[12];
    const float* Wl2  = (const float*)d_in[13];
    const float* bl2  = (const float*)d_in[14];
    const float* Wr2  = (const float*)d_in[15];
    float* out = (float*)d_out;

    char* ws = (char*)d_ws;
    float* h1   = (float*)(ws + H1_OFF);
    float* agg0 = (float*)(ws + AGG0_OFF);
    float* agg1 = (float*)(ws + AGG1_OFF);
    float* agg2 = (float*)(ws + AGG2_OFF);
    float* h2   = (float*)(ws + H2_OFF);
    float* cnt  = (float*)(ws + CNT_OFF);

    // ------------------ layer 0: (512000,100) -> (102400,256) ------------
    {
        int naggs = N_DST0 * IN_C;
        sage_zero_f32<<<divup(naggs, 256), 256, 0, stream>>>(agg0, naggs);
        sage_zero_f32<<<divup(N_DST0, 256), 256, 0, stream>>>(cnt, N_DST0);
        sage_scatter<<<divup((long long)E0 * 32, 256), 256, 0, stream>>>(
            x, src0, dst0, agg0, cnt, E0, IN_C);
        sage_meandiv<<<divup(naggs, 256), 256, 0, stream>>>(agg0, cnt, naggs, IN_C);
        int tilesM = N_DST0 / 16, groupsN = (HID_C / 16) / 8;        // 6400 x 2
        sage_gemm_wmma<8, false>
            <<<divup((long long)tilesM * groupsN * 32, 256), 256, 0, stream>>>(
            agg0, x, Wl0, Wr0, bl0, h1, N_DST0, IN_C, HID_C, groupsN, 1);
    }
    // ------------------ layer 1: (102400,256) -> (10240,256) -------------
    {
        int naggs = N_DST1 * HID_C;
        sage_zero_f32<<<divup(naggs, 256), 256, 0, stream>>>(agg1, naggs);
        sage_zero_f32<<<divup(N_DST1, 256), 256, 0, stream>>>(cnt, N_DST1);
        sage_scatter<<<divup((long long)E1 * 32, 256), 256, 0, stream>>>(
            h1, src1, dst1, agg1, cnt, E1, HID_C);
        sage_meandiv<<<divup(naggs, 256), 256, 0, stream>>>(agg1, cnt, naggs, HID_C);
        int tilesM = N_DST1 / 16, groupsN = (HID_C / 16) / 8;        // 640 x 2
        sage_gemm_wmma<8, false>
            <<<divup((long long)tilesM * groupsN * 32, 256), 256, 0, stream>>>(
            agg1, h1, Wl1, Wr1, bl1, h2, N_DST1, HID_C, HID_C, groupsN, 1);
    }
    // ------------------ layer 2: (10240,256) -> (1024,47) + logsoftmax ---
    {
        int naggs = N_DST2 * HID_C;
        sage_zero_f32<<<divup(naggs, 256), 256, 0, stream>>>(agg2, naggs);
        sage_zero_f32<<<divup(N_DST2, 256), 256, 0, stream>>>(cnt, N_DST2);
        sage_scatter<<<divup((long long)E2 * 32, 256), 256, 0, stream>>>(
            h2, src2, dst2, agg2, cnt, E2, HID_C);
        sage_meandiv<<<divup(naggs, 256), 256, 0, stream>>>(agg2, cnt, naggs, HID_C);
        int tilesM = N_DST2 / 16, groupsN = 1;                       // 64 x 1 (3 tiles)
        sage_gemm_wmma<3, true>
            <<<divup((long long)tilesM * groupsN * 32, 256), 256, 0, stream>>>(
            agg2, h2, Wl2, Wr2, bl2, out, N_DST2, HID_C, OUT_C, groupsN, 0);
        sage_logsoftmax<<<divup((long long)N_DST2 * 32, 256), 256, 0, stream>>>(
            out, N_DST2, OUT_C);
    }
}